// Kitnet_35742717837416
// MI455X (gfx1250) — compile-verified
//
#include <hip/hip_runtime.h>
#include <hip/hip_bf16.h>
#include <math.h>

typedef __attribute__((ext_vector_type(2))) float v2f;
typedef __attribute__((ext_vector_type(4))) float v4f;
typedef __attribute__((ext_vector_type(8))) float v8f;

#define INPUT_SIZE   1000
#define M_FEAT       10
#define NCLUST       100
#define H_SMALL      5
#define H_META       50

#define SAMPLES_PER_BLOCK 32
#define CHUNK        8
#define NCHUNKS      (SAMPLES_PER_BLOCK / CHUNK)
#define BLOCK_THREADS 128

// Strides chosen so WMMA A-matrix fetch patterns are LDS bank-conflict-free
#define XS_STRIDE    1012   // 1012 % 64 = 52 -> distinct banks across the 8 staged rows
#define ERR_STRIDE   101    // odd -> 16 distinct banks for M=0..15 A fetches
#define HM_STRIDE    53     // odd, >= 52 so K-pad columns 50..51 exist (pre-zeroed)
#define ER_STRIDE    101

#define K1_CHUNKS    25     // GEMM1: K = 100
#define K2_CHUNKS    13     // GEMM2: K = 50 padded to 52

__global__ __launch_bounds__(BLOCK_THREADS)
void kitnet_fused_wmma(const float* __restrict__ x,
                       const int*   __restrict__ feat_idx,
                       const float* __restrict__ W_enc,
                       const float* __restrict__ b_enc,
                       const float* __restrict__ W_dec,
                       const float* __restrict__ b_dec,
                       const float* __restrict__ We1,
                       const float* __restrict__ be1,
                       const float* __restrict__ Wd1,
                       const float* __restrict__ bd1,
                       float* __restrict__ out,
                       int batch)
{
    __shared__ float xs   [CHUNK * XS_STRIDE];               // staged x rows (32.4 KB)
    __shared__ float err_s[SAMPLES_PER_BLOCK * ERR_STRIDE];  // per-cluster RMSE (12.9 KB)
    __shared__ float hm_s [SAMPLES_PER_BLOCK * HM_STRIDE];   // meta hidden, K-padded (6.8 KB)
    __shared__ float er_s [SAMPLES_PER_BLOCK * ER_STRIDE];   // meta reconstruction (12.9 KB)

    const int tid        = threadIdx.x;
    const int lane       = tid & 31;
    const int wid        = tid >> 5;
    const int block_base = blockIdx.x * SAMPLES_PER_BLOCK;

    // Pre-zero hm_s K-pad columns (50, 51) so GEMM2 A-reads of the pad are exact zeros.
    if (tid < 2 * SAMPLES_PER_BLOCK)
        hm_s[(tid >> 1) * HM_STRIDE + H_META + (tid & 1)] = 0.0f;

    // ---------------- Phase 1: gather + 100 tiny AEs -> err_s ----------------
    for (int chunk = 0; chunk < NCHUNKS; ++chunk) {
        const int sbase = block_base + chunk * CHUNK;

        // Coalesced non-temporal float4 staging (x is read exactly once device-wide),
        // plus prefetch of the next chunk's lines into L2.
        for (int i = tid; i < CHUNK * (INPUT_SIZE / 4); i += BLOCK_THREADS) {
            const int s  = i / (INPUT_SIZE / 4);
            const int f4 = i % (INPUT_SIZE / 4);
            const int gs = sbase + s;
            if (gs < batch) {
                const v4f v = __builtin_nontemporal_load(
                    (const v4f*)(x + (size_t)gs * INPUT_SIZE) + f4);
                float* dst = &xs[s * XS_STRIDE + f4 * 4];
                dst[0] = v[0]; dst[1] = v[1]; dst[2] = v[2]; dst[3] = v[3];
                if (chunk + 1 < NCHUNKS && gs + CHUNK < batch) {
                    __builtin_prefetch(
                        x + (size_t)(gs + CHUNK) * INPUT_SIZE + f4 * 4, 0, 0);
                }
            }
        }
        __syncthreads();

        // One thread per (sample, cluster): 10 -> 5 -> 10 AE, RMSE over m
        for (int task = tid; task < CHUNK * NCLUST; task += BLOCK_THREADS) {
            const int s = task / NCLUST;   // lanes in a wave mostly share s -> random-bank gather
            const int c = task % NCLUST;

            float xg[M_FEAT];
            #pragma unroll
            for (int m = 0; m < M_FEAT; ++m)
                xg[m] = xs[s * XS_STRIDE + feat_idx[c * M_FEAT + m]];

            float h[H_SMALL];
            #pragma unroll
            for (int j = 0; j < H_SMALL; ++j) {
                float t = b_enc[c * H_SMALL + j];
                #pragma unroll
                for (int m = 0; m < M_FEAT; ++m)
                    t = fmaf(xg[m], W_enc[(c * M_FEAT + m) * H_SMALL + j], t);
                h[j] = fmaxf(t, 0.0f);
            }

            float acc = 0.0f;
            #pragma unroll
            for (int m = 0; m < M_FEAT; ++m) {
                float r = b_dec[c * M_FEAT + m];
                #pragma unroll
                for (int j = 0; j < H_SMALL; ++j)
                    r = fmaf(h[j], W_dec[(c * H_SMALL + j) * M_FEAT + m], r);
                const float d = xg[m] - r;
                acc = fmaf(d, d, acc);
            }
            err_s[(chunk * CHUNK + s) * ERR_STRIDE + c] =
                sqrtf(acc * (1.0f / (float)M_FEAT));
        }
        __syncthreads();   // also protects xs reuse next chunk
    }

    // ---------------- Phase 2: meta AE via V_WMMA_F32_16X16X4_F32 ----------------
    // Wave -> (sample tile of 16, N-tile parity). All WMMA control flow wave-uniform.
    const int tile   = wid & 1;           // sample tile 0/1
    const int half   = wid >> 1;          // N-tile parity
    const int ncol_l = lane & 15;
    const int ksel   = (lane < 16) ? 0 : 2;    // A/B K-pair select per half-wave
    const int srow0  = tile * 16 + ncol_l;     // A-matrix row (M = lane%16)
    const int rowoff = (lane < 16) ? 0 : 8;    // D-layout M offset

    // GEMM1: hm = relu(err(16x100) @ We1(100x50) + be1), N padded to 64.
    // Padded output columns are discarded at the guarded store, so B only needs a
    // clamped (in-bounds) address -- no selects, no divergent branches.
    for (int nt = half; nt < 4; nt += 2) {
        const int  ncol = nt * 16 + ncol_l;
        const bool okn  = (ncol < H_META);
        const int  nc   = okn ? ncol : (H_META - 1);

        v2f breg[K1_CHUNKS];                   // stage all B up-front: one batched wait
        #pragma unroll
        for (int i = 0; i < K1_CHUNKS; ++i) {
            const int k = 4 * i + ksel;
            breg[i][0] = We1[(k)     * H_META + nc];
            breg[i][1] = We1[(k + 1) * H_META + nc];
        }

        v8f acc = {0.f, 0.f, 0.f, 0.f, 0.f, 0.f, 0.f, 0.f};
        #pragma unroll
        for (int i = 0; i < K1_CHUNKS; ++i) {
            v2f a;
            a[0] = err_s[srow0 * ERR_STRIDE + 4 * i + ksel];
            a[1] = err_s[srow0 * ERR_STRIDE + 4 * i + ksel + 1];
            acc = __builtin_amdgcn_wmma_f32_16x16x4_f32(
                false, a, false, breg[i], (short)0, acc, false, false);
        }

        const float bias = okn ? be1[nc] : 0.0f;
        #pragma unroll
        for (int r = 0; r < 8; ++r) {
            const float v    = fmaxf(acc[r] + bias, 0.0f);
            const int   srow = tile * 16 + r + rowoff;
            if (okn) hm_s[srow * HM_STRIDE + ncol] = v;
        }
    }
    __syncthreads();

    // GEMM2: err_rec = hm(16x50) @ Wd1(50x100) + bd1, K padded to 52 (A pad rows are
    // exact zeros in hm_s, so clamped B k-addresses contribute 0*finite = 0 exactly).
    for (int nt = half; nt < 7; nt += 2) {
        const int  ncol = nt * 16 + ncol_l;
        const bool okn  = (ncol < NCLUST);
        const int  nc   = okn ? ncol : (NCLUST - 1);

        v2f breg[K2_CHUNKS];
        #pragma unroll
        for (int i = 0; i < K2_CHUNKS; ++i) {
            const int k0  = 4 * i + ksel;
            const int kb0 = (k0     < H_META) ? k0     : (H_META - 1);
            const int kb1 = (k0 + 1 < H_META) ? k0 + 1 : (H_META - 1);
            breg[i][0] = Wd1[kb0 * NCLUST + nc];
            breg[i][1] = Wd1[kb1 * NCLUST + nc];
        }

        v8f acc = {0.f, 0.f, 0.f, 0.f, 0.f, 0.f, 0.f, 0.f};
        #pragma unroll
        for (int i = 0; i < K2_CHUNKS; ++i) {
            v2f a;
            a[0] = hm_s[srow0 * HM_STRIDE + 4 * i + ksel];
            a[1] = hm_s[srow0 * HM_STRIDE + 4 * i + ksel + 1];
            acc = __builtin_amdgcn_wmma_f32_16x16x4_f32(
                false, a, false, breg[i], (short)0, acc, false, false);
        }

        const float bias = okn ? bd1[nc] : 0.0f;
        #pragma unroll
        for (int r = 0; r < 8; ++r) {
            const int srow = tile * 16 + r + rowoff;
            if (okn) er_s[srow * ER_STRIDE + ncol] = acc[r] + bias;
        }
    }
    __syncthreads();

    // ---------------- Phase 3: final RMSE + sigmoid, coalesced store ----------------
    if (tid < SAMPLES_PER_BLOCK) {
        const int s  = tid;
        const int gs = block_base + s;
        if (gs < batch) {
            float acc = 0.0f;
            for (int c = 0; c < NCLUST; ++c) {
                const float d = err_s[s * ERR_STRIDE + c] - er_s[s * ER_STRIDE + c];
                acc = fmaf(d, d, acc);
            }
            const float fe = sqrtf(acc * (1.0f / (float)NCLUST));
            out[gs] = 1.0f / (1.0f + __expf(-fe));
        }
    }
}

extern "C" void kernel_launch(void* const* d_in, const int* in_sizes, int n_in,
                              void* d_out, int out_size, void* d_ws, size_t ws_size,
                              hipStream_t stream) {
    const float* x        = (const float*)d_in[0];
    const int*   feat_idx = (const int*)  d_in[1];
    const float* W_enc    = (const float*)d_in[2];
    const float* b_enc    = (const float*)d_in[3];
    const float* W_dec    = (const float*)d_in[4];
    const float* b_dec    = (const float*)d_in[5];
    const float* We1      = (const float*)d_in[6];
    const float* be1      = (const float*)d_in[7];
    const float* Wd1      = (const float*)d_in[8];
    const float* bd1      = (const float*)d_in[9];
    float* out = (float*)d_out;

    const int batch = in_sizes[0] / INPUT_SIZE;
    const int grid  = (batch + SAMPLES_PER_BLOCK - 1) / SAMPLES_PER_BLOCK;

    kitnet_fused_wmma<<<grid, BLOCK_THREADS, 0, stream>>>(
        x, feat_idx, W_enc, b_enc, W_dec, b_dec, We1, be1, Wd1, bd1, out, batch);
}